// Full_67568425501232
// MI455X (gfx1250) — compile-verified
//
#include <hip/hip_runtime.h>

typedef float v2f __attribute__((ext_vector_type(2)));
typedef float v8f __attribute__((ext_vector_type(8)));

#define HP1    33
#define KROWS  1089          // 33*33 valid K rows per (i,j) slice
#define K2PAD  1092          // padded to multiple of 4
#define NSLICE 1089          // (i,j) slice count
#define CHUNKS 4
#define NUNITS (NSLICE * CHUNKS)
#define TSLICE 34848         // 33*33*32 floats per contiguous (i,j) slice of T
#define WAVES_PER_BLK 8
#define MAX_NWG 544

// out[b,o] = sum_K A'[b,K] * T_flat[K,o];  A'[b, s*1089+t] = w01[b,s] * A2[b,t]
__global__ __launch_bounds__(256) void full_main(const float* __restrict__ nh,
                                                 const float* __restrict__ T,
                                                 float* __restrict__ partial,
                                                 int totalWaves)
{
    // 139776 + 4224 + 4224 = 148224 B of LDS (<=160KB -> 2 workgroups/WGP)
    __shared__ float ldsA2[32 * K2PAD];   // A2[b][t] = haug(b,2,t/33)*haug(b,3,t%33), zero-padded
    __shared__ float ldsA0[32 * HP1];     // haug(b,0,i)
    __shared__ float ldsA1[32 * HP1];     // haug(b,1,j)

    const int tid = threadIdx.x;

    for (int idx = tid; idx < 32 * HP1; idx += 256) {
        int b = idx / HP1, i = idx % HP1;
        ldsA0[idx] = (i < 32) ? nh[(b * 4 + 0) * 32 + i] : 1.0f;
        ldsA1[idx] = (i < 32) ? nh[(b * 4 + 1) * 32 + i] : 1.0f;
    }
    for (int idx = tid; idx < 32 * K2PAD; idx += 256) {
        int b = idx / K2PAD, t = idx % K2PAD;
        float v = 0.0f;
        if (t < KROWS) {
            int kk = t / HP1, l = t % HP1;
            float h2 = (kk < 32) ? nh[(b * 4 + 2) * 32 + kk] : 1.0f;
            float h3 = (l  < 32) ? nh[(b * 4 + 3) * 32 + l ] : 1.0f;
            v = h2 * h3;
        }
        ldsA2[idx] = v;
    }
    __syncthreads();

    const int lane = tid & 31;
    const int wv   = tid >> 5;
    // force wave-uniformity so the work loop is a scalar loop (EXEC stays all-1s
    // through every WMMA, and no per-lane exec-mask bookkeeping)
    const int gwave      = __builtin_amdgcn_readfirstlane(blockIdx.x * WAVES_PER_BLK + wv);
    const int strideWave = __builtin_amdgcn_readfirstlane(totalWaves);
    const int half = lane >> 4;   // K-pair half per ISA A/B layout
    const int nlo  = lane & 15;

    const int ldsRowLo = nlo * K2PAD;          // A2 row base, m-tile 0
    const int ldsRowHi = (16 + nlo) * K2PAD;   // A2 row base, m-tile 1

    v8f acc00 = {}, acc01 = {}, acc10 = {}, acc11 = {};

    for (int u = gwave; u < NUNITS; u += strideWave) {
        const int s = u >> 2;
        const int c = u & 3;
        const int tstart = c * 272;           // chunks cover rows [c*272, c*272+272)
        const int si = s / HP1, sj = s % HP1;

        // fold w01[b,s] into the A operand (rows m = nlo and 16+nlo)
        const float w0 = ldsA0[nlo * HP1 + si]        * ldsA1[nlo * HP1 + sj];
        const float w1 = ldsA0[(16 + nlo) * HP1 + si] * ldsA1[(16 + nlo) * HP1 + sj];

        const float* __restrict__ Ts = T + (size_t)s * TSLICE;

        // 68 unclamped K-steps (rows tstart .. tstart+271, max 1087: always in range)
        int t = tstart;
#pragma unroll 2
        for (int step = 0; step < 68; ++step, t += 4) {
            const int ka = t + half * 2;
            v2f a0p = *(const v2f*)&ldsA2[ldsRowLo + ka];
            v2f a1p = *(const v2f*)&ldsA2[ldsRowHi + ka];
            v2f A0; A0.x = a0p.x * w0; A0.y = a0p.y * w0;
            v2f A1; A1.x = a1p.x * w1; A1.y = a1p.y * w1;

            // B 4x16: rows ka, ka+1 ; cols nlo / nlo+16
            const float* rp = Ts + (size_t)ka * 32 + nlo;
            v2f B0, B1;
            B0.x = rp[0];   B0.y = rp[32];
            B1.x = rp[16];  B1.y = rp[48];

            acc00 = __builtin_amdgcn_wmma_f32_16x16x4_f32(false, A0, false, B0, (short)0, acc00, false, false);
            acc01 = __builtin_amdgcn_wmma_f32_16x16x4_f32(false, A0, false, B1, (short)0, acc01, false, false);
            acc10 = __builtin_amdgcn_wmma_f32_16x16x4_f32(false, A1, false, B0, (short)0, acc10, false, false);
            acc11 = __builtin_amdgcn_wmma_f32_16x16x4_f32(false, A1, false, B1, (short)0, acc11, false, false);
        }

        // chunk 3 epilogue: one clamped step for rows 1088..1091
        // (ldsA2 is zero for t>=1089, so clamped/duplicated B rows contribute 0)
        if (c == 3) {
            const int ka = 1088 + half * 2;            // 1088 or 1090
            v2f a0p = *(const v2f*)&ldsA2[ldsRowLo + ka];
            v2f a1p = *(const v2f*)&ldsA2[ldsRowHi + ka];
            v2f A0; A0.x = a0p.x * w0; A0.y = a0p.y * w0;
            v2f A1; A1.x = a1p.x * w1; A1.y = a1p.y * w1;

            const float* rp = Ts + (size_t)(KROWS - 1) * 32 + nlo;  // row 1088 (clamped)
            v2f B0, B1;
            B0.x = rp[0];  B0.y = rp[0];
            B1.x = rp[16]; B1.y = rp[16];

            acc00 = __builtin_amdgcn_wmma_f32_16x16x4_f32(false, A0, false, B0, (short)0, acc00, false, false);
            acc01 = __builtin_amdgcn_wmma_f32_16x16x4_f32(false, A0, false, B1, (short)0, acc01, false, false);
            acc10 = __builtin_amdgcn_wmma_f32_16x16x4_f32(false, A1, false, B0, (short)0, acc10, false, false);
            acc11 = __builtin_amdgcn_wmma_f32_16x16x4_f32(false, A1, false, B1, (short)0, acc11, false, false);
        }
    }

    // intra-block reduction through LDS (reuse the A2 region; 8*1024 floats fit)
    __syncthreads();
    float* red = ldsA2;
    for (int r = 0; r < 8; ++r) {
        // C/D layout: vgpr r holds M = r (lanes 0-15) / M = 8+r (lanes 16-31)
        int m0 = half * 8 + r;        // m-tile 0
        int m1 = 16 + half * 8 + r;   // m-tile 1
        red[wv * 1024 + m0 * 32 + nlo]      = acc00[r];
        red[wv * 1024 + m0 * 32 + nlo + 16] = acc01[r];
        red[wv * 1024 + m1 * 32 + nlo]      = acc10[r];
        red[wv * 1024 + m1 * 32 + nlo + 16] = acc11[r];
    }
    __syncthreads();
    for (int idx = tid; idx < 1024; idx += 256) {
        float sum = 0.0f;
        for (int w = 0; w < WAVES_PER_BLK; ++w) sum += red[w * 1024 + idx];
        partial[(size_t)blockIdx.x * 1024 + idx] = sum;
    }
}

__global__ __launch_bounds__(256) void full_reduce(const float* __restrict__ partial,
                                                   float* __restrict__ out, int nwg)
{
    int o = blockIdx.x * 256 + threadIdx.x;
    if (o >= 1024) return;
    float s = 0.0f;
    for (int w = 0; w < nwg; ++w) s += partial[(size_t)w * 1024 + o];
    out[o] = s;
}

extern "C" void kernel_launch(void* const* d_in, const int* in_sizes, int n_in,
                              void* d_out, int out_size, void* d_ws, size_t ws_size,
                              hipStream_t stream) {
    const float* nh = (const float*)d_in[0];   // [32,4,32] fp32
    const float* T  = (const float*)d_in[1];   // [33,33,33,33,32] fp32
    float* out      = (float*)d_out;           // [32,32] fp32
    float* partial  = (float*)d_ws;

    int nwg = MAX_NWG;
    size_t need = (size_t)nwg * 1024 * sizeof(float);
    if (ws_size < need) {
        nwg = (int)(ws_size / (1024 * sizeof(float)));
        if (nwg < 1) nwg = 1;
    }

    full_main<<<nwg, 256, 0, stream>>>(nh, T, partial, nwg * WAVES_PER_BLK);
    full_reduce<<<4, 256, 0, stream>>>(partial, out, nwg);
}